// ProjectionSimToPointCloud_4501125726269
// MI455X (gfx1250) — compile-verified
//
#include <hip/hip_runtime.h>
#include <hip/hip_bf16.h>

// ProjectionSimToPointCloud: range image (H,W,4) -> point cloud (H*W, 3).
// Pure streaming / bandwidth-bound kernel:
//   read 256 MiB, write 192 MiB  ->  ~20 us roofline at 23.3 TB/s HBM.
// No matrix math exists in this op, so no WMMA; we instead exploit CDNA5
// cache controls (non-temporal TH on loads/stores) and the hardware
// sin/cos units with pre-normalized (revolutions) arguments.

#define IMG_H 2048
#define IMG_W 8192
#define LOG2_W 13

// pitch_rev(i) = pitch/(2*pi) = 1/24 - i/24576   (fov +/-15 deg)
#define PITCH_A 0.041666666666666664f      // 1/24
#define PITCH_B 4.0690104166666666e-05f    // 1/24576
// yaw_rev(j) = yaw/(2*pi) = j/8192 - 0.5
#define INV_W 1.220703125e-04f             // 1/8192 (exact power of two)

// Clang-native vector type (accepted by __builtin_nontemporal_store,
// lowers to one global_store_b128 with NT temporal hint).
typedef float v4f __attribute__((ext_vector_type(4)));

__global__ __launch_bounds__(256) void
ProjectionSimToPointCloud_4501125726269_kernel(const float* __restrict__ img,
                                               float* __restrict__ out,
                                               int npix) {
    const int tid = blockIdx.x * blockDim.x + threadIdx.x;
    const int p   = tid << 2;                 // 4 consecutive pixels per lane
    if (p >= npix) return;

    // Row / column of the first pixel (W = 8192, power of two).
    const float i_f = (float)(p >> LOG2_W);
    const float j_f = (float)(p & (IMG_W - 1));

    // sin(pitch) is constant across the 4 pixels (same row).
    const float pitch_rev = fmaf(i_f, -PITCH_B, PITCH_A);
    const float sp = __builtin_amdgcn_sinf(pitch_rev);   // v_sin_f32

    float x[4], y[4], z[4];
#pragma unroll
    for (int k = 0; k < 4; ++k) {
        // depth = image[i, j+k, 3]; non-temporal streaming read.
        const float depth =
            __builtin_nontemporal_load(img + (size_t)(p + k) * 4 + 3);

        // yaw in revolutions: j/W - 0.5 (exact), feed HW sin/cos directly.
        const float yaw_rev = fmaf(j_f + (float)k, INV_W, -0.5f);
        const float sy = __builtin_amdgcn_sinf(yaw_rev); // v_sin_f32
        const float cy = __builtin_amdgcn_cosf(yaw_rev); // v_cos_f32

        x[k] = depth * cy;
        y[k] = -depth * sy;
        z[k] = depth * sp;
    }

    // 4 points * 12 B = 48 B contiguous, 16B-aligned per lane:
    // pack into three b128 non-temporal stores.
    v4f* o4 = (v4f*)(out + (size_t)p * 3);
    v4f v0 = {x[0], y[0], z[0], x[1]};
    v4f v1 = {y[1], z[1], x[2], y[2]};
    v4f v2 = {z[2], x[3], y[3], z[3]};
    __builtin_nontemporal_store(v0, o4 + 0);
    __builtin_nontemporal_store(v1, o4 + 1);
    __builtin_nontemporal_store(v2, o4 + 2);
}

extern "C" void kernel_launch(void* const* d_in, const int* in_sizes, int n_in,
                              void* d_out, int out_size, void* d_ws, size_t ws_size,
                              hipStream_t stream) {
    const float* img = (const float*)d_in[0];
    float* out = (float*)d_out;

    const int npix    = in_sizes[0] / 4;        // 2048*8192 = 16,777,216
    const int threads = (npix + 3) / 4;         // 4 pixels per thread
    const int block   = 256;                    // 8 wave32s
    const int grid    = (threads + block - 1) / block;

    ProjectionSimToPointCloud_4501125726269_kernel<<<grid, block, 0, stream>>>(
        img, out, npix);
}